// agg_corr_27376121544785
// MI455X (gfx1250) — compile-verified
//
#include <hip/hip_runtime.h>
#include <stdint.h>

#define H 72
#define W 72
#define PLANE (H * W)        // 5184
#define NWELEM (9 * PLANE)   // 46656 weight floats -> 186,624 B LDS
#define T_X1 9               // x1 planes handled per block (72 / 9 = 8 groups)
#define BLOCK 512            // 16 wave32 waves

#define AS1 __attribute__((address_space(1)))
#define AS3 __attribute__((address_space(3)))

typedef int v4i __attribute__((vector_size(4 * sizeof(int))));

#if defined(__has_builtin)
#if __has_builtin(__builtin_amdgcn_global_load_async_to_lds_b128)
#define USE_ASYNC_LDS 1
#endif
#if __has_builtin(__builtin_amdgcn_s_wait_asynccnt)
#define HAVE_WAIT_ASYNC_BUILTIN 1
#endif
#endif

static __device__ __forceinline__ int imax0(int a) { return a < 0 ? 0 : a; }
static __device__ __forceinline__ int imin(int a, int b) { return a < b ? a : b; }

extern "C" __global__ __launch_bounds__(BLOCK) void agg_corr_kernel(
    const float* __restrict__ corr, const float* __restrict__ weights,
    float* __restrict__ out) {
  extern __shared__ float w_lds[];  // NWELEM floats (dynamic LDS, 186,624 B)
  const int tid = threadIdx.x;

  // ---- Stage all 9*5184 weights into LDS (gfx1250 async-to-LDS path) ----
  {
#if USE_ASYNC_LDS
    AS1 v4i* gsrc = (AS1 v4i*)weights;           // global src, v4i32-typed
    AS3 v4i* ldst = (AS3 v4i*)&w_lds[0];         // LDS dst, v4i32-typed
    for (int i = tid; i < NWELEM / 4; i += BLOCK) {
      __builtin_amdgcn_global_load_async_to_lds_b128(gsrc + i, ldst + i, 0, 0);
    }
#if HAVE_WAIT_ASYNC_BUILTIN
    __builtin_amdgcn_s_wait_asynccnt(0);
#else
    asm volatile("s_wait_asynccnt 0" ::: "memory");
#endif
#else
    const float4* src = (const float4*)weights;
    float4* dst = (float4*)&w_lds[0];
    for (int i = tid; i < NWELEM / 4; i += BLOCK) dst[i] = src[i];
#endif
  }
  __syncthreads();

  const int y1 = blockIdx.y;          // [0, 72)
  const int x1g = blockIdx.x * T_X1;  // start of this block's x1 group

  const int ym1 = imax0(y1 - 1);
  const int yp1 = imin(y1 + 1, H - 1);

  for (int xi = 0; xi < T_X1; ++xi) {
    const int x1 = x1g + xi;
    const int xm1 = imax0(x1 - 1);
    const int xp1 = imin(x1 + 1, W - 1);

    // Plane base for direction d = (clamp(y1 - sy_d), clamp(x1 - sx_d)).
    // SHIFTS order: (0,1)(-1,1)(-1,0)(-1,-1)(0,0)(0,-1)(1,-1)(1,0)(1,1)
    const size_t b0 = (size_t)(ym1 * W + x1) * PLANE;   // ( 0, 1)
    const size_t b1 = (size_t)(ym1 * W + xp1) * PLANE;  // (-1, 1)
    const size_t b2 = (size_t)(y1 * W + xp1) * PLANE;   // (-1, 0)
    const size_t b3 = (size_t)(yp1 * W + xp1) * PLANE;  // (-1,-1)
    const size_t b4 = (size_t)(y1 * W + x1) * PLANE;    // ( 0, 0)
    const size_t b5 = (size_t)(yp1 * W + x1) * PLANE;   // ( 0,-1)
    const size_t b6 = (size_t)(yp1 * W + xm1) * PLANE;  // ( 1,-1)
    const size_t b7 = (size_t)(y1 * W + xm1) * PLANE;   // ( 1, 0)
    const size_t b8 = (size_t)(ym1 * W + xm1) * PLANE;  // ( 1, 1)

    float* outp = out + (size_t)(y1 * W + x1) * PLANE;

    for (int e = tid; e < PLANE; e += BLOCK) {
      const int y2 = e / W;
      const int x2 = e - y2 * W;
      const int rm = imax0(y2 - 1) * W;  // clamp(y2-1) row
      const int r0 = y2 * W;
      const int rp = imin(y2 + 1, H - 1) * W;
      const int cm = imax0(x2 - 1);  // clamp(x2-1) col
      const int c0 = x2;
      const int cp = imin(x2 + 1, W - 1);

      float acc;
      acc = corr[b0 + (size_t)(rm + c0)] * w_lds[0 * PLANE + e];
      acc = fmaf(corr[b1 + (size_t)(rm + cp)], w_lds[1 * PLANE + e], acc);
      acc = fmaf(corr[b2 + (size_t)(r0 + cp)], w_lds[2 * PLANE + e], acc);
      acc = fmaf(corr[b3 + (size_t)(rp + cp)], w_lds[3 * PLANE + e], acc);
      acc = fmaf(corr[b4 + (size_t)(r0 + c0)], w_lds[4 * PLANE + e], acc);
      acc = fmaf(corr[b5 + (size_t)(rp + c0)], w_lds[5 * PLANE + e], acc);
      acc = fmaf(corr[b6 + (size_t)(rp + cm)], w_lds[6 * PLANE + e], acc);
      acc = fmaf(corr[b7 + (size_t)(r0 + cm)], w_lds[7 * PLANE + e], acc);
      acc = fmaf(corr[b8 + (size_t)(rm + cm)], w_lds[8 * PLANE + e], acc);

      // Output is write-once, never re-read: non-temporal store keeps
      // corr+weights resident in the 192 MB L2.
      __builtin_nontemporal_store(acc, &outp[e]);
    }
  }
}

extern "C" void kernel_launch(void* const* d_in, const int* in_sizes, int n_in,
                              void* d_out, int out_size, void* d_ws,
                              size_t ws_size, hipStream_t stream) {
  const float* corr = (const float*)d_in[0];     // (1, 5184, 72, 72) f32
  const float* weights = (const float*)d_in[1];  // (1, 9, 5184) f32
  float* out = (float*)d_out;                    // (1, 5184, 72, 72) f32

  dim3 grid(W / T_X1, H);  // 8 x 72 = 576 blocks
  agg_corr_kernel<<<grid, dim3(BLOCK), NWELEM * sizeof(float), stream>>>(
      corr, weights, out);
}